// ImageCut_79542794322035
// MI455X (gfx1250) — compile-verified
//
#include <hip/hip_runtime.h>
#include <math.h>

typedef __attribute__((ext_vector_type(2))) float v2f;
typedef __attribute__((ext_vector_type(8))) float v8f;
typedef __attribute__((ext_vector_type(4))) int   v4i;

#define IMG     256
#define FM_C    512
#define FM_HW   256      // 16*16 spatial
#define REMAIN  115
#define CROP    230

#define AS1 __attribute__((address_space(1)))
#define AS3 __attribute__((address_space(3)))

#if defined(__gfx1250__) && __has_builtin(__builtin_amdgcn_global_load_async_to_lds_b128)
#define USE_ASYNC_LDS_B128 1
#elif defined(__gfx1250__) && __has_builtin(__builtin_amdgcn_global_load_async_to_lds_b32)
#define USE_ASYNC_LDS_B32 1
#endif

__device__ __forceinline__ void wait_async_then_barrier() {
#if __has_builtin(__builtin_amdgcn_s_wait_asynccnt)
    __builtin_amdgcn_s_wait_asynccnt(0);
#else
    asm volatile("s_wait_asynccnt 0" ::: "memory");
#endif
    __syncthreads();
}

__device__ __forceinline__ int cut_start(int i) {
    if (i <= REMAIN) return 0;
    if ((IMG - i) <= REMAIN) return IMG - CROP;
    return i - REMAIN;
}

// ---------------------------------------------------------------------------
// Kernel 2 (bulk of the traffic): bilinear crop-resize, one block per output
// row. The two shared source rows (2 KB) are staged into LDS with async
// global->LDS b128 copies (two wave-instructions per row), then each thread
// does the separable blend and one coalesced b32 store.
// ---------------------------------------------------------------------------
__launch_bounds__(256)
__global__ void ImageCut_bilinear(const float* __restrict__ x,
                                  const int* __restrict__ offs,
                                  float* __restrict__ out) {
    __shared__ float row0[IMG];
    __shared__ float row1[IMG];

    const int tid  = threadIdx.x;          // output column
    const int orow = blockIdx.x & 255;     // output row
    const int ch   = (blockIdx.x >> 8) % 3;
    const int b    = blockIdx.x / (3 * 256);

    const int x_left = offs[2 * b];
    const int y_top  = offs[2 * b + 1];

    const float scale = (float)(CROP - 1) / (float)(IMG - 1);

    float sr  = (float)orow * scale;
    int   i0r = (int)sr;                   // sr >= 0 -> trunc == floor
    float wr  = sr - (float)i0r;
    int   i1r = (i0r + 1 < CROP) ? i0r + 1 : CROP - 1;

    const float* base  = x + ((size_t)b * 3 + ch) * (size_t)(IMG * IMG);
    const float* prow0 = base + (size_t)(x_left + i0r) * IMG;
    const float* prow1 = base + (size_t)(x_left + i1r) * IMG;

#if defined(USE_ASYNC_LDS_B128)
    if (tid < 128) {                       // waves 0-3: 64 lanes x 16B per row
        const int e = (tid & 63) * 4;
        const float* src = (tid < 64) ? (prow0 + e) : (prow1 + e);
        float*       dst = (tid < 64) ? (&row0[e]) : (&row1[e]);
        __builtin_amdgcn_global_load_async_to_lds_b128((AS1 v4i*)src,
                                                       (AS3 v4i*)dst, 0, 0);
    }
    wait_async_then_barrier();
#elif defined(USE_ASYNC_LDS_B32)
    __builtin_amdgcn_global_load_async_to_lds_b32((AS1 int*)(prow0 + tid),
                                                  (AS3 int*)(&row0[tid]), 0, 0);
    __builtin_amdgcn_global_load_async_to_lds_b32((AS1 int*)(prow1 + tid),
                                                  (AS3 int*)(&row1[tid]), 0, 0);
    wait_async_then_barrier();
#else
    row0[tid] = prow0[tid];
    row1[tid] = prow1[tid];
    __syncthreads();
#endif

    float sc  = (float)tid * scale;
    int   i0c = (int)sc;
    float wc  = sc - (float)i0c;
    int   i1c = (i0c + 1 < CROP) ? i0c + 1 : CROP - 1;
    const int c0 = y_top + i0c;
    const int c1 = y_top + i1c;

    float left  = row0[c0] * (1.0f - wr) + row1[c0] * wr;
    float right = row0[c1] * (1.0f - wr) + row1[c1] * wr;
    out[(size_t)blockIdx.x * IMG + tid] = left * (1.0f - wc) + right * wc;
}

// ---------------------------------------------------------------------------
// Kernel 1: per-batch CAM + argmax -> crop offsets (one 256-thread block / batch)
//   Stage 1: fw[c] = spatial mean of feature_map, computed on the matrix pipe:
//            D = A(16ch x 4pos, f32) x ones(4x16) chained over K  => row sums.
//   Stage 2: cam[p] = sum_c relu(fw[c]*fm[c,p])  (coalesced: thread == position)
//   Stage 3: argmax with first-index tie-break, emit (x_left, y_top).
// ---------------------------------------------------------------------------
__launch_bounds__(256)
__global__ void ImageCut_cam_argmax(const float* __restrict__ fm,
                                    int* __restrict__ offs) {
    __shared__ float fw[FM_C];
    __shared__ float camv[256];
    __shared__ int   cami[256];

    const int b     = blockIdx.x;
    const int tid   = threadIdx.x;
    const int wave  = tid >> 5;
    const int lane  = tid & 31;
    const int m     = lane & 15;   // A-matrix row (channel within tile)
    const int khalf = lane >> 4;   // lanes 16-31 hold K=2,3

    const float* fmb = fm + (size_t)b * FM_C * FM_HW;

    // ---- Stage 1: channel means via v_wmma_f32_16x16x4_f32 ----
    v2f ones; ones[0] = 1.0f; ones[1] = 1.0f;
    for (int tt = 0; tt < 4; ++tt) {             // 8 waves * 4 tiles = 512 ch
        const int c0 = wave * 64 + tt * 16;
        const float* ap = fmb + (size_t)(c0 + m) * FM_HW + 2 * khalf;
        v8f acc = {};
        for (int k0 = 0; k0 < FM_HW; k0 += 4) {
            v2f a = *(const v2f*)(ap + k0);      // A[m][k..k+1] for this lane
            acc = __builtin_amdgcn_wmma_f32_16x16x4_f32(
                false, a, false, ones, (short)0, acc, false, false);
        }
        // D layout: vgpr v, lanes 0-15 -> M=v ; lanes 16-31 -> M=8+v (any N column)
        if (lane == 0) {
#pragma unroll
            for (int v = 0; v < 8; ++v) fw[c0 + v] = acc[v] * (1.0f / 256.0f);
        } else if (lane == 16) {
#pragma unroll
            for (int v = 0; v < 8; ++v) fw[c0 + 8 + v] = acc[v] * (1.0f / 256.0f);
        }
    }
    __syncthreads();

    // ---- Stage 2: cam (coalesced across spatial positions) ----
    float acc = 0.0f;
    const float* col = fmb + tid;
    for (int c = 0; c < FM_C; ++c)
        acc += fmaxf(fw[c] * col[(size_t)c * FM_HW], 0.0f);
    camv[tid] = acc;
    cami[tid] = tid;
    __syncthreads();

    // ---- Stage 3: argmax (first index wins ties) ----
    for (int s = 128; s > 0; s >>= 1) {
        if (tid < s) {
            float v1 = camv[tid], v2 = camv[tid + s];
            int   i1 = cami[tid], i2 = cami[tid + s];
            if (v2 > v1 || (v2 == v1 && i2 < i1)) { camv[tid] = v2; cami[tid] = i2; }
        }
        __syncthreads();
    }
    if (tid == 0) {
        int idx = cami[0];
        int ix  = 16 * (idx >> 4);   // IMG*fx/16
        int iy  = 16 * (idx & 15);
        offs[2 * b]     = cut_start(ix);
        offs[2 * b + 1] = cut_start(iy);
    }
}

extern "C" void kernel_launch(void* const* d_in, const int* in_sizes, int n_in,
                              void* d_out, int out_size, void* d_ws, size_t ws_size,
                              hipStream_t stream) {
    const float* x  = (const float*)d_in[0];   // (128, 3, 256, 256) f32
    const float* fm = (const float*)d_in[1];   // (128, 512, 16, 16) f32
    float* out = (float*)d_out;                // (128, 3, 256, 256) f32
    int*   offs = (int*)d_ws;                  // 128 * 2 ints

    const int B = 128;
    ImageCut_cam_argmax<<<B, 256, 0, stream>>>(fm, offs);
    ImageCut_bilinear<<<B * 3 * 256, 256, 0, stream>>>(x, offs, out);
}